// CorrectJAmbiguityBlock_70205535421263
// MI455X (gfx1250) — compile-verified
//
#include <hip/hip_runtime.h>

namespace {
constexpr int kB = 4;
constexpr int kN = 512;                  // nodes / block-columns per row
constexpr int kD = 3 * kN;               // 1536
constexpr int kRowsPerBlock = 4;
constexpr int kThreads = 256;            // 8 wave32
// cells per block = kRowsPerBlock * kN = 2048 -> 8 cells per thread
constexpr int kIters = (kRowsPerBlock * kN) / kThreads;   // 8
}

typedef int v2i __attribute__((ext_vector_type(2)));
typedef __attribute__((address_space(1))) v2i* as1_v2i_ptr;
typedef __attribute__((address_space(3))) v2i* as3_v2i_ptr;

__global__ __launch_bounds__(kThreads)
void jflip_stream_kernel(const float* __restrict__ H,
                         const float* __restrict__ u_s,
                         float* __restrict__ out) {
    __shared__ __align__(16) float s_u[kN];

    const int t    = threadIdx.x;
    const int b    = blockIdx.y;
    const int row0 = blockIdx.x * kRowsPerBlock;

    // ---- Stage u_s[b][0..N) into LDS via the gfx1250 async global->LDS path.
    // 256 threads x 8 bytes = 2048 B = 512 floats, one b64 async op per lane.
#if __has_builtin(__builtin_amdgcn_global_load_async_to_lds_b64)
    {
        const float* g = u_s + (size_t)b * kN + 2 * t;
        __builtin_amdgcn_global_load_async_to_lds_b64(
            (as1_v2i_ptr)(g),
            (as3_v2i_ptr)(&s_u[2 * t]),
            /*offset=*/0, /*cpol=*/0);
#if __has_builtin(__builtin_amdgcn_s_wait_asynccnt)
        __builtin_amdgcn_s_wait_asynccnt(0);
#else
        asm volatile("s_wait_asynccnt 0" ::: "memory");
#endif
    }
#else
    ((float2*)s_u)[t] = ((const float2*)(u_s + (size_t)b * kN))[t];
#endif
    __syncthreads();

    // Convert staged u_s values to +-1 signs in place (saves per-cell compares).
    {
        const float a0 = s_u[2 * t];
        const float a1 = s_u[2 * t + 1];
        s_u[2 * t]     = (a0 < 0.0f) ? -1.0f : 1.0f;
        s_u[2 * t + 1] = (a1 < 0.0f) ? -1.0f : 1.0f;
    }
    __syncthreads();

    // ---- Streaming pass: one 3-float block-cell per thread per iteration. ----
#pragma unroll
    for (int it = 0; it < kIters; ++it) {
        const int rl = it >> 1;                 // compile-time row-in-block 0..3
        const int q  = t + (it & 1) * kThreads; // block column 0..511 (vector)
        const int R  = row0 + rl;               // global row (scalar)
        const int p  = R / 3;                   // block row (scalar)
        const int r  = R - 3 * p;               // local row 0..2 (scalar)

        const size_t off = ((size_t)(b * kD + R) * kN + q) * 3;  // float offset
        const float* hp = H + off;
        const float h0 = hp[0];
        const float h1 = hp[1];
        const float h2 = hp[2];

        const int   m    = (q > p) ? q : p;     // sign source: larger node index
        const float sgn  = s_u[m];              // +-1
        const bool  zero = (q == p);            // diagonal 3x3 block -> 0
        const bool  r2   = (r == 2);            // scalar condition

        // factor for local cols 0,1 (c!=2) and local col 2 (c==2):
        float fA = r2 ? sgn  : 1.0f;
        float fB = r2 ? 1.0f : sgn;
        fA = zero ? 0.0f : fA;
        fB = zero ? 0.0f : fB;

        float* op = out + off;
        op[0] = h0 * fA;
        op[1] = h1 * fA;
        op[2] = h2 * fB;
    }
}

extern "C" void kernel_launch(void* const* d_in, const int* in_sizes, int n_in,
                              void* d_out, int out_size, void* d_ws, size_t ws_size,
                              hipStream_t stream) {
    // setup_inputs order: H (f32), u_s (f32), gather_idx, gather_idx2, u_s_gather_idx
    const float* H  = (const float*)d_in[0];
    const float* us = (const float*)d_in[1];
    float* out = (float*)d_out;
    (void)in_sizes; (void)n_in; (void)out_size; (void)d_ws; (void)ws_size;

    dim3 grid(kD / kRowsPerBlock, kB);   // (384, 4)
    jflip_stream_kernel<<<grid, kThreads, 0, stream>>>(H, us, out);
}